// TimeSeriesRegister_27135603376581
// MI455X (gfx1250) — compile-verified
//
#include <hip/hip_runtime.h>
#include <hip/hip_bf16.h>

// Problem constants (match reference)
#define B_  8192
#define S_  512
#define F_  64
#define R_  4096
#define D_  256
#define T_  16

typedef __attribute__((ext_vector_type(2))) float v2f;
typedef __attribute__((ext_vector_type(8))) float v8f;

// ---------------------------------------------------------------------------
// Kernel 0: zero the loss accumulator (last element of d_out)
// ---------------------------------------------------------------------------
__global__ void zero_loss_kernel(float* __restrict__ loss) {
    if (threadIdx.x == 0) *loss = 0.0f;
}

// ---------------------------------------------------------------------------
// Kernel 1: temporal mean pool + projection (bandwidth-dominant: 1 GiB read).
// b128 loads: thread t handles float4 column group fq = t&15, s-group sg = t>>4.
// Each wave requests 512 contiguous bytes per global_load_b128.
// ---------------------------------------------------------------------------
__global__ void mean_project_kernel(const float* __restrict__ x,     // [B,S,F]
                                    const float* __restrict__ W,     // [D,F]
                                    const float* __restrict__ bias,  // [D]
                                    float* __restrict__ xe) {        // [B,D]
    __shared__ float  ldsW[F_ * 257];   // transposed+padded: ldsW[j*257 + d]
    __shared__ float4 red4[256];
    __shared__ float  xmean[F_];

    const int b = blockIdx.x;
    const int t = threadIdx.x;

    // stage W transposed (b128 global reads; 4 consecutive j of one row d)
    {
        const float4* W4 = (const float4*)W;
        for (int i = t; i < (D_ * F_) / 4; i += 256) {
            const float4 w = W4[i];
            const int base = i * 4;
            const int d0 = base >> 6;
            const int j0 = base & 63;
            ldsW[(j0    ) * 257 + d0] = w.x;
            ldsW[(j0 + 1) * 257 + d0] = w.y;
            ldsW[(j0 + 2) * 257 + d0] = w.z;
            ldsW[(j0 + 3) * 257 + d0] = w.w;
        }
    }

    // mean over S with float4 loads
    const int fq = t & 15;    // float4 column group (covers f = 4*fq .. 4*fq+3)
    const int sg = t >> 4;    // s-group 0..15
    const float4* xb4 = (const float4*)(x + (size_t)b * (S_ * F_));
    float ax = 0.f, ay = 0.f, az = 0.f, aw = 0.f;
    for (int s = sg; s < S_; s += 16) {
        const float4 v = xb4[s * (F_ / 4) + fq];
        ax += v.x; ay += v.y; az += v.z; aw += v.w;
    }
    red4[t] = make_float4(ax, ay, az, aw);
    __syncthreads();

    if (t < 16) {
        float mx = 0.f, my = 0.f, mz = 0.f, mw = 0.f;
        for (int g = 0; g < 16; ++g) {
            const float4 v = red4[g * 16 + t];
            mx += v.x; my += v.y; mz += v.z; mw += v.w;
        }
        const float inv = 1.0f / (float)S_;
        xmean[t * 4 + 0] = mx * inv;
        xmean[t * 4 + 1] = my * inv;
        xmean[t * 4 + 2] = mz * inv;
        xmean[t * 4 + 3] = mw * inv;
    }
    __syncthreads();

    // projection: thread t == output dim d (conflict-free LDS reads, stride 257)
    float a = bias[t];
#pragma unroll 16
    for (int j = 0; j < F_; ++j)
        a += xmean[j] * ldsW[j * 257 + t];
    xe[(size_t)b * D_ + t] = a;
}

// ---------------------------------------------------------------------------
// Kernel 2: per-codebook-row squared norm (b128 loads)
// ---------------------------------------------------------------------------
__global__ void reg_sq_kernel(const float* __restrict__ reg,   // [R,D]
                              float* __restrict__ reg_sq) {    // [R]
    const int r = blockIdx.x * blockDim.x + threadIdx.x;
    if (r < R_) {
        const float4* p = (const float4*)(reg + (size_t)r * D_);
        float acc = 0.0f;
#pragma unroll 8
        for (int j = 0; j < D_ / 4; ++j) {
            const float4 v = p[j];
            acc += v.x * v.x + v.y * v.y + v.z * v.z + v.w * v.w;
        }
        reg_sq[r] = acc;
    }
}

// ---------------------------------------------------------------------------
// Kernel 3: WMMA distance GEMM + fused argmin (unchanged core).
// Grid: B/16 blocks x 128 threads (4 waves). Each wave: 16-row xe tile vs 64
// N-tiles of the codebook, K=256 in 64 v_wmma_f32_16x16x4_f32 steps.
// argmin key = reg_sq[r] - 2*score (monotone equivalent of the distance).
// ---------------------------------------------------------------------------
#define LDSA 258   // padded row stride (floats) for the A tile

__global__ void dist_argmin_kernel(const float* __restrict__ xe,      // [B,D]
                                   const float* __restrict__ reg,     // [R,D]
                                   const float* __restrict__ reg_sq,  // [R]
                                   int* __restrict__ closest) {       // [B]
    __shared__ float ldsA[16 * LDSA];
    __shared__ float lds_min[4 * 16 * 16];
    __shared__ int   lds_idx[4 * 16 * 16];

    const int btile = blockIdx.x;          // 0..511
    const int tid   = threadIdx.x;         // 0..127
    const int wave  = tid >> 5;
    const int lane  = tid & 31;
    const int lr    = lane & 15;
    const int hi    = lane >> 4;           // half-wave select

    // stage the 16x256 A tile (xe rows) into LDS, coalesced
    for (int i = tid; i < 16 * D_; i += 128) {
        const int row = i >> 8;
        const int col = i & 255;
        ldsA[row * LDSA + col] = xe[((size_t)btile * 16 + row) * D_ + col];
    }
    __syncthreads();

    float best[8];
    int   bidx[8];
#pragma unroll
    for (int i = 0; i < 8; ++i) { best[i] = 3.4e38f; bidx[i] = 0; }

    // A-fragment base (ISA 16x4 f32 layout: lanes 0-15 K={0,1}, lanes 16-31 K={2,3})
    const float* Arow = ldsA + lr * LDSA + 2 * hi;

    const int nt0 = wave * 64;
    for (int nt = nt0; nt < nt0 + 64; ++nt) {
        const int n = nt * 16 + lr;                        // codebook row (lane's column)
        const float* Brow = reg + (size_t)n * D_ + 2 * hi; // B[k][n] = register[n][k]

        // prefetch next tile's codebook row for this lane (global_prefetch)
        if (nt + 1 < nt0 + 64)
            __builtin_prefetch(reg + (size_t)(n + 16) * D_, 0, 0);

        v8f c = {};
#pragma unroll 8
        for (int k0 = 0; k0 < D_; k0 += 4) {
            const v2f a  = *(const v2f*)(Arow + k0);   // ds_load_b64
            const v2f bv = *(const v2f*)(Brow + k0);   // global_load_b64 (L2-resident)
            c = __builtin_amdgcn_wmma_f32_16x16x4_f32(
                    false, a, false, bv, (short)0, c, false, false);
        }

        const float rs = reg_sq[n];
#pragma unroll
        for (int i = 0; i < 8; ++i) {
            // C layout: VGPR i -> row (i + 8*hi), col n
            const float key = rs - 2.0f * c[i];
            if (key < best[i]) { best[i] = key; bidx[i] = n; }
        }
    }

    // per-lane partials -> LDS (row = i + 8*hi, 16 column-lanes per row per wave)
#pragma unroll
    for (int i = 0; i < 8; ++i) {
        const int row = i + 8 * hi;
        lds_min[(wave * 16 + row) * 16 + lr] = best[i];
        lds_idx[(wave * 16 + row) * 16 + lr] = bidx[i];
    }
    __syncthreads();

    // final reduce: thread `row` scans 4 waves x 16 lanes, first-index tie-break
    if (tid < 16) {
        float m = 3.4e38f; int mi = 0x7fffffff;
        for (int w = 0; w < 4; ++w)
            for (int l = 0; l < 16; ++l) {
                const float v  = lds_min[(w * 16 + tid) * 16 + l];
                const int   id = lds_idx[(w * 16 + tid) * 16 + l];
                if (v < m || (v == m && id < mi)) { m = v; mi = id; }
            }
        closest[btile * 16 + tid] = mi;
    }
}

// ---------------------------------------------------------------------------
// Kernel 4: gather selected codebook row, tiled b128 output writes, loss.
// Thread t: float4 column group c4 = t&63, quarter q = t>>6 writes rows
// q, q+4, q+8, q+12 (each wave-store = 512 contiguous bytes).
// ---------------------------------------------------------------------------
__global__ void finalize_kernel(const float* __restrict__ xe,      // [B,D]
                                const float* __restrict__ reg,     // [R,D]
                                const int* __restrict__ closest,   // [B]
                                float* __restrict__ out,           // [B,T,D]
                                float* __restrict__ loss) {
    __shared__ float red[256];
    const int b  = blockIdx.x;
    const int t  = threadIdx.x;
    const int c  = closest[b];
    const int c4 = t & 63;
    const int q  = t >> 6;

    const float4 sel = ((const float4*)(reg + (size_t)c * D_))[c4];

    if (q == 0) {
        const float4 xv = ((const float4*)(xe + (size_t)b * D_))[c4];
        const float dx = xv.x - sel.x, dy = xv.y - sel.y;
        const float dz = xv.z - sel.z, dw = xv.w - sel.w;
        red[t] = dx * dx + dy * dy + dz * dz + dw * dw;
    } else {
        red[t] = 0.0f;
    }

    float4* ob4 = (float4*)(out + (size_t)b * (T_ * D_));
#pragma unroll
    for (int k = 0; k < 4; ++k)
        ob4[(q + 4 * k) * (D_ / 4) + c4] = sel;

    __syncthreads();
    for (int off = 128; off > 0; off >>= 1) {
        if (t < off) red[t] += red[t + off];
        __syncthreads();
    }
    if (t == 0) atomicAdd(loss, red[0] * (1.0f / (float)B_));
}

// ---------------------------------------------------------------------------
extern "C" void kernel_launch(void* const* d_in, const int* in_sizes, int n_in,
                              void* d_out, int out_size, void* d_ws, size_t ws_size,
                              hipStream_t stream) {
    const float* x    = (const float*)d_in[0];   // [B,S,F]
    const float* W    = (const float*)d_in[1];   // [D,F]
    const float* bias = (const float*)d_in[2];   // [D]
    const float* reg  = (const float*)d_in[3];   // [R,D]

    float* out     = (float*)d_out;                          // Xd flat, then loss
    float* xe      = (float*)d_ws;                           // B*D floats (8 MB)
    float* reg_sq  = xe + (size_t)B_ * D_;                   // R floats
    int*   closest = (int*)(reg_sq + R_);                    // B ints
    float* loss    = out + (size_t)B_ * T_ * D_;             // scalar slot

    zero_loss_kernel<<<1, 32, 0, stream>>>(loss);
    mean_project_kernel<<<B_, 256, 0, stream>>>(x, W, bias, xe);
    reg_sq_kernel<<<R_ / 256, 256, 0, stream>>>(reg, reg_sq);
    dist_argmin_kernel<<<B_ / 16, 128, 0, stream>>>(xe, reg, reg_sq, closest);
    finalize_kernel<<<B_, 256, 0, stream>>>(xe, reg, closest, out, loss);
}